// neural_network_58342835749015
// MI455X (gfx1250) — compile-verified
//
#include <hip/hip_runtime.h>
#include <cstdint>
#include <math.h>

typedef float v2f __attribute__((ext_vector_type(2)));
typedef float v8f __attribute__((ext_vector_type(8)));

#define BATCH  512
#define TSTEPS 2048
#define FEAT   32
#define NCOH   16   // COH_SIZE = 8*2
#define NBULK  24   // BULK_SIZE = 8*3
#define NOUT   6

// ---- material constants (match reference D_NP / DINV_NP analytically) ----
#define D0c   0.01f
#define DFc   0.1f
#define EMODc 1000.0f
#define NUc   0.3f
#define SIGY0c 10.0f
#define HARDc  100.0f
#define GMODc (EMODc / (2.0f * (1.0f + NUc)))            // shear modulus
#define CCc   (EMODc / (1.0f - NUc * NUc))               // plane-stress factor
#define C11c  (CCc)
#define C12c  (CCc * NUc)
#define C33c  (CCc * (1.0f - NUc) * 0.5f)
#define I11c  (1.0f / EMODc)
#define I12c  (-NUc / EMODc)
#define I33c  (1.0f / C33c)

// =====================================================================
// Phase 1: projection GEMM  proj = x(1M x 32) @ [W11;W12]^T  via
// V_WMMA_F32_16X16X4_F32 (exact fp32). Each wave: one 16-row tile,
// 3 n-tiles (coh 0..15 | bulk 0..15 | bulk 16..23 + pad), 8 k-steps.
// Output staged transposed: coh[t][b][16], bulk[t][b][24] for a
// coalesced, mostly-L2-resident scan phase.
// =====================================================================
__global__ __launch_bounds__(256) void proj_wmma_kernel(
    const float* __restrict__ x, const float* __restrict__ W11,
    const float* __restrict__ W12, float* __restrict__ coh,
    float* __restrict__ bulk)
{
  const int lane = threadIdx.x & 31;
  const int wave = threadIdx.x >> 5;
  const int m  = lane & 15;   // M index for A, N index for B, col for D
  const int hi = lane >> 4;   // half-wave: selects K pair / D row offset

  const long tile = (long)blockIdx.x * 8 + wave;     // 65536 tiles total
  const long row0 = tile * 16;                       // flat row = b*T + t
  const int  b    = (int)(row0 >> 11);               // /2048
  const int  t0   = (int)(row0 & 2047);              // %2048 (tile stays in one b)

  // --- A fragments: lane m holds row row0+m; v0/v1 = K{0,1} (lo) / K{2,3} (hi)
  v2f a[8];
  const float* xr = x + (row0 + m) * FEAT + 2 * hi;  // column base 0 or 2
#pragma unroll
  for (int kk = 0; kk < 8; ++kk) {
    const float* p = xr + kk * 4;
    v2f av; av.x = __builtin_nontemporal_load(p);
            av.y = __builtin_nontemporal_load(p + 1);
    a[kk] = av;
  }

  // --- B fragments: lane m = output channel n; K pair selected by hi.
  v2f bw0[8], bw1[8], bw2[8];
#pragma unroll
  for (int kk = 0; kk < 8; ++kk) {
    const int col = kk * 4 + 2 * hi;
    v2f v;
    v.x = W11[m * FEAT + col];     v.y = W11[m * FEAT + col + 1];
    bw0[kk] = v;
    v.x = W12[m * FEAT + col];     v.y = W12[m * FEAT + col + 1];
    bw1[kk] = v;
    if (m < 8) { v.x = W12[(16 + m) * FEAT + col]; v.y = W12[(16 + m) * FEAT + col + 1]; }
    else       { v.x = 0.0f; v.y = 0.0f; }           // pad columns 40..47
    bw2[kk] = v;
  }

  v8f acc0 = {}, acc1 = {}, acc2 = {};
#pragma unroll
  for (int kk = 0; kk < 8; ++kk) {
    acc0 = __builtin_amdgcn_wmma_f32_16x16x4_f32(false, a[kk], false, bw0[kk],
                                                 (short)0, acc0, false, false);
    acc1 = __builtin_amdgcn_wmma_f32_16x16x4_f32(false, a[kk], false, bw1[kk],
                                                 (short)0, acc1, false, false);
    acc2 = __builtin_amdgcn_wmma_f32_16x16x4_f32(false, a[kk], false, bw2[kk],
                                                 (short)0, acc2, false, false);
  }

  // --- store D: element g is (row M = g + 8*hi  -> t = t0+M, col = m)
#pragma unroll
  for (int g = 0; g < 8; ++g) {
    const long t     = t0 + g + 8 * hi;
    const long baseC = (t * (long)BATCH + b) * NCOH;
    const long baseB = (t * (long)BATCH + b) * NBULK;
    coh[baseC + m]  = acc0[g];
    bulk[baseB + m] = acc1[g];
    if (m < 8) bulk[baseB + 16 + m] = acc2[g];
  }
}

// =====================================================================
// Per-step recurrence math for one (b, p) chain.
// =====================================================================
__device__ __forceinline__ void step_math(
    float dn, float ds, float e0v, float e1v, float e2v,
    float& d_hist, float& ep0, float& ep1, float& ep2, float& kap,
    const float (&w)[6][3], float (&pout)[6])
{
  // cohesive damage
  const float dnp = fmaxf(dn, 0.0f);
  const float lam = sqrtf(dnp * dnp + ds * ds + 1e-12f);
  float d_new = DFc * (lam - D0c) / (lam * (DFc - D0c));
  d_new = fminf(fmaxf(d_new, 0.0f), 1.0f);
  const bool loading = (d_new > d_hist);
  const float d = fmaxf(d_hist, d_new);
  d_hist = d;

  // bulk trial stress (plane stress)
  const float x0 = e0v - ep0, x1 = e1v - ep1, x2 = e2v - ep2;
  const float sxx = C11c * x0 + C12c * x1;
  const float syy = C12c * x0 + C11c * x1;
  const float sxy = C33c * x2;
  const float seq = sqrtf(sxx * sxx - sxx * syy + syy * syy + 3.0f * sxy * sxy + 1e-12f);
  const float sigY = SIGY0c + HARDc * kap;
  const float dk = fmaxf(seq - sigY, 0.0f) / (3.0f * GMODc + HARDc);
  const float f = (seq > sigY) ? (sigY + HARDc * dk) / seq : 1.0f;
  const float s0 = f * sxx, s1 = f * syy, s2 = f * sxy;
  ep0 = e0v - (I11c * s0 + I12c * s1);
  ep1 = e1v - (I12c * s0 + I11c * s1);
  ep2 = e2v - I33c * s2;
  kap += dk;

  // damage applied to output stress only (not to plastic history)
  const float scale = loading ? 1.0f : (1.0f - d);
  const float o0 = s0 * scale, o1 = s1 * scale, o2 = s2 * scale;
#pragma unroll
  for (int o = 0; o < 6; ++o)
    pout[o] = w[o][0] * o0 + w[o][1] * o1 + w[o][2] * o2;
}

__device__ __forceinline__ float softplusf(float v) {
  return fmaxf(v, 0.0f) + log1pf(expf(-fabsf(v)));
}

// =====================================================================
// Phase 2: sequential scan. One thread per (b, p) chain: 4096 threads.
// Threads ordered b*8+p, so the 8 p's of a batch element occupy an
// aligned 8-lane group -> shfl_xor butterfly for the W2 contraction.
// Software-pipelined loads hide memory latency behind the dependent
// math chain.
// =====================================================================
__global__ __launch_bounds__(64) void scan_kernel(
    const float* __restrict__ coh, const float* __restrict__ bulk,
    const float* __restrict__ W2, float* __restrict__ out)
{
  const int tid = blockIdx.x * 64 + threadIdx.x;
  const int b = tid >> 3;
  const int p = tid & 7;

  // softplus(W2) slice needed by this point: columns 3p..3p+2, all 6 rows
  float w[6][3];
#pragma unroll
  for (int o = 0; o < 6; ++o)
#pragma unroll
    for (int j = 0; j < 3; ++j)
      w[o][j] = softplusf(W2[o * NBULK + 3 * p + j]);

  float d_hist = 0.0f, ep0 = 0.0f, ep1 = 0.0f, ep2 = 0.0f, kap = 0.0f;

  const size_t cstride = (size_t)BATCH * NCOH;
  const size_t bstride = (size_t)BATCH * NBULK;
  const size_t cbase = (size_t)b * NCOH + 2 * p;
  const size_t bbase = (size_t)b * NBULK + 3 * p;
  float* obase = out + (size_t)b * TSTEPS * NOUT;

  float2 de = *(const float2*)(coh + cbase);
  float e0v = bulk[bbase + 0], e1v = bulk[bbase + 1], e2v = bulk[bbase + 2];

  for (int t = 0; t < TSTEPS; ++t) {
    float2 de_n = de;
    float e0n = e0v, e1n = e1v, e2n = e2v;
    if (t + 1 < TSTEPS) {                       // prefetch next step
      const size_t cc = cbase + (size_t)(t + 1) * cstride;
      const size_t bb = bbase + (size_t)(t + 1) * bstride;
      de_n = *(const float2*)(coh + cc);
      e0n = bulk[bb + 0]; e1n = bulk[bb + 1]; e2n = bulk[bb + 2];
    }

    float pout[6];
    step_math(de.x, de.y, e0v, e1v, e2v, d_hist, ep0, ep1, ep2, kap, w, pout);

    // reduce over the 8 points of this batch element (intra-wave)
#pragma unroll
    for (int o = 0; o < 6; ++o) {
      pout[o] += __shfl_xor(pout[o], 1, 8);
      pout[o] += __shfl_xor(pout[o], 2, 8);
      pout[o] += __shfl_xor(pout[o], 4, 8);
    }
    if (p == 0) {
      float* op = obase + (size_t)t * NOUT;
      op[0] = pout[0]; op[1] = pout[1]; op[2] = pout[2];
      op[3] = pout[3]; op[4] = pout[4]; op[5] = pout[5];
    }

    de = de_n; e0v = e0n; e1v = e1n; e2v = e2n;
  }
}

// =====================================================================
// Fallback: fused scan (projections computed inline) if the workspace
// cannot hold the staged projections. Correct but no WMMA path.
// =====================================================================
__global__ __launch_bounds__(64) void scan_fused_kernel(
    const float* __restrict__ x, const float* __restrict__ W11,
    const float* __restrict__ W12, const float* __restrict__ W2,
    float* __restrict__ out)
{
  const int tid = blockIdx.x * 64 + threadIdx.x;
  const int b = tid >> 3;
  const int p = tid & 7;

  float wa[FEAT], wb[FEAT], wc0[FEAT], wc1[FEAT], wc2[FEAT];
#pragma unroll
  for (int f = 0; f < FEAT; ++f) {
    wa[f]  = W11[(2 * p + 0) * FEAT + f];
    wb[f]  = W11[(2 * p + 1) * FEAT + f];
    wc0[f] = W12[(3 * p + 0) * FEAT + f];
    wc1[f] = W12[(3 * p + 1) * FEAT + f];
    wc2[f] = W12[(3 * p + 2) * FEAT + f];
  }
  float w[6][3];
#pragma unroll
  for (int o = 0; o < 6; ++o)
#pragma unroll
    for (int j = 0; j < 3; ++j)
      w[o][j] = softplusf(W2[o * NBULK + 3 * p + j]);

  float d_hist = 0.0f, ep0 = 0.0f, ep1 = 0.0f, ep2 = 0.0f, kap = 0.0f;
  const float* xb = x + (size_t)b * TSTEPS * FEAT;
  float* obase = out + (size_t)b * TSTEPS * NOUT;

  for (int t = 0; t < TSTEPS; ++t) {
    const float* xr = xb + (size_t)t * FEAT;
    float dn = 0, ds = 0, e0v = 0, e1v = 0, e2v = 0;
#pragma unroll
    for (int f = 0; f < FEAT; ++f) {
      const float xv = xr[f];
      dn  = fmaf(wa[f], xv, dn);
      ds  = fmaf(wb[f], xv, ds);
      e0v = fmaf(wc0[f], xv, e0v);
      e1v = fmaf(wc1[f], xv, e1v);
      e2v = fmaf(wc2[f], xv, e2v);
    }
    float pout[6];
    step_math(dn, ds, e0v, e1v, e2v, d_hist, ep0, ep1, ep2, kap, w, pout);
#pragma unroll
    for (int o = 0; o < 6; ++o) {
      pout[o] += __shfl_xor(pout[o], 1, 8);
      pout[o] += __shfl_xor(pout[o], 2, 8);
      pout[o] += __shfl_xor(pout[o], 4, 8);
    }
    if (p == 0) {
      float* op = obase + (size_t)t * NOUT;
      op[0] = pout[0]; op[1] = pout[1]; op[2] = pout[2];
      op[3] = pout[3]; op[4] = pout[4]; op[5] = pout[5];
    }
  }
}

extern "C" void kernel_launch(void* const* d_in, const int* in_sizes, int n_in,
                              void* d_out, int out_size, void* d_ws, size_t ws_size,
                              hipStream_t stream) {
  const float* x   = (const float*)d_in[0];   // (512, 2048, 32)
  const float* W11 = (const float*)d_in[1];   // (16, 32)
  const float* W12 = (const float*)d_in[2];   // (24, 32)
  const float* W2  = (const float*)d_in[3];   // (6, 24)
  float* out = (float*)d_out;                 // (512, 2048, 6)

  const size_t cohElems  = (size_t)TSTEPS * BATCH * NCOH;
  const size_t bulkElems = (size_t)TSTEPS * BATCH * NBULK;
  const size_t needBytes = (cohElems + bulkElems) * sizeof(float); // ~168 MB

  if (ws_size >= needBytes) {
    float* coh  = (float*)d_ws;
    float* bulk = coh + cohElems;
    // 65536 16-row tiles, 8 waves (tiles) per 256-thread block
    proj_wmma_kernel<<<8192, 256, 0, stream>>>(x, W11, W12, coh, bulk);
    scan_kernel<<<64, 64, 0, stream>>>(coh, bulk, W2, out);
  } else {
    scan_fused_kernel<<<64, 64, 0, stream>>>(x, W11, W12, W2, out);
  }
}